// SparseLinearKAN_10325101380247
// MI455X (gfx1250) — compile-verified
//
#include <hip/hip_runtime.h>
#include <stdint.h>
#include <stddef.h>

#define BATCH 256
#define WAVES_PER_BLOCK 8
#define DEPTH 4      // LDS slots per wave (1KB each)
#define INFLIGHT 3   // rows in flight (2 async instructions each)

#if defined(__has_builtin)
#if __has_builtin(__builtin_amdgcn_global_load_async_to_lds_b128) && \
    __has_builtin(__builtin_amdgcn_s_wait_asynccnt)
#define USE_ASYNC_LDS 1
#endif
#endif

#define AS1 __attribute__((address_space(1)))
#define AS3 __attribute__((address_space(3)))
typedef int v4i __attribute__((vector_size(16)));   // matches builtin param type

// ---------------- CSR build phase ----------------

__global__ void sk_zero_i32(int* __restrict__ p, int n) {
  int i = blockIdx.x * blockDim.x + threadIdx.x;
  if (i < n) p[i] = 0;
}

__global__ void sk_hist_rows(const long long* __restrict__ rows,
                             int* __restrict__ counts, int nnz) {
  int i = blockIdx.x * blockDim.x + threadIdx.x;
  if (i < nnz) atomicAdd(&counts[(int)rows[i]], 1);
}

// single-block exclusive scan over `n` counts -> row_ptr[n+1], cursor[n]
__global__ __launch_bounds__(1024) void sk_exscan(const int* __restrict__ counts,
                                                  int* __restrict__ row_ptr,
                                                  int* __restrict__ cursor, int n) {
  __shared__ int part[1024];
  __shared__ int pre[1024];
  const int t = threadIdx.x;
  const int chunk = (n + 1023) >> 10;
  int beg = t * chunk;
  int end = beg + chunk; if (end > n) end = n;
  int s = 0;
  for (int i = beg; i < end; ++i) s += counts[i];
  part[t] = s;
  __syncthreads();
  if (t == 0) {
    int run = 0;
    for (int i = 0; i < 1024; ++i) { pre[i] = run; run += part[i]; }
    row_ptr[n] = run;
  }
  __syncthreads();
  int run = pre[t];
  for (int i = beg; i < end; ++i) {
    row_ptr[i] = run;
    cursor[i]  = run;
    run += counts[i];
  }
}

__global__ void sk_scatter(const long long* __restrict__ rows,
                           const long long* __restrict__ colsIn,
                           const float* __restrict__ valsIn,
                           int* __restrict__ cursor,
                           int* __restrict__ colsOut,
                           float* __restrict__ valsOut, int nnz) {
  int i = blockIdx.x * blockDim.x + threadIdx.x;
  if (i < nnz) {
    int r = (int)rows[i];
    int p = atomicAdd(&cursor[r], 1);
    colsOut[p] = (int)colsIn[i];
    valsOut[p] = valsIn[i];
  }
}

// ---------------- SpMM phase: one wave32 per output row ----------------

__global__ __launch_bounds__(256) void sk_spmm_csr(
    const float* __restrict__ x, const int* __restrict__ row_ptr,
    const int* __restrict__ cols, const float* __restrict__ vals,
    const float* __restrict__ bias, float* __restrict__ out, int nrows) {
  const int lane = threadIdx.x & 31;
  const int wave = threadIdx.x >> 5;
  const int row  = blockIdx.x * WAVES_PER_BLOCK + wave;
  if (row >= nrows) return;
  const int p0 = row_ptr[row];
  const int n  = row_ptr[row + 1] - p0;

  float4 acc0 = make_float4(0.f, 0.f, 0.f, 0.f);
  float4 acc1 = make_float4(0.f, 0.f, 0.f, 0.f);

#ifdef USE_ASYNC_LDS
  // 8 waves * 4 slots * 1KB = 32KB LDS per workgroup
  __shared__ float lds[WAVES_PER_BLOCK * DEPTH * BATCH];
  float* slots = &lds[wave * (DEPTH * BATCH)];

  int issued = (n < INFLIGHT) ? n : INFLIGHT;
  for (int j = 0; j < issued; ++j) {
    const float* g = x + (size_t)cols[p0 + j] * BATCH + lane * 4;  // 16B per lane
    float* l = slots + (j & (DEPTH - 1)) * BATCH + lane * 4;
    // inst-offset is added to BOTH global and LDS addresses (ISA 15.18)
    __builtin_amdgcn_global_load_async_to_lds_b128((AS1 v4i*)g, (AS3 v4i*)l, 0,   0);
    __builtin_amdgcn_global_load_async_to_lds_b128((AS1 v4i*)g, (AS3 v4i*)l, 512, 0);
  }

  for (int k = 0; k < n; ++k) {
    // in-order completion: asynccnt <= 2*(rows still allowed in flight)
    const int d = issued - k - 1;
    if (d >= 2)      __builtin_amdgcn_s_wait_asynccnt(4);
    else if (d == 1) __builtin_amdgcn_s_wait_asynccnt(2);
    else             __builtin_amdgcn_s_wait_asynccnt(0);
    asm volatile("" ::: "memory");  // keep the LDS reads below the wait

    const float  v  = vals[p0 + k];
    const float4* sp = (const float4*)(slots + (k & (DEPTH - 1)) * BATCH + lane * 8);
    float4 a = sp[0];
    float4 b = sp[1];
    acc0.x += v * a.x; acc0.y += v * a.y; acc0.z += v * a.z; acc0.w += v * a.w;
    acc1.x += v * b.x; acc1.y += v * b.y; acc1.z += v * b.z; acc1.w += v * b.w;

    if (issued < n) {
      // slot being refilled must not have outstanding ds_loads against it
      asm volatile("s_wait_dscnt 0" ::: "memory");
      const float* g = x + (size_t)cols[p0 + issued] * BATCH + lane * 4;
      float* l = slots + (issued & (DEPTH - 1)) * BATCH + lane * 4;
      __builtin_amdgcn_global_load_async_to_lds_b128((AS1 v4i*)g, (AS3 v4i*)l, 0,   0);
      __builtin_amdgcn_global_load_async_to_lds_b128((AS1 v4i*)g, (AS3 v4i*)l, 512, 0);
      ++issued;
    }
  }
#else
  // fallback: direct-to-VGPR gather (compiler pipelines via loadcnt)
  for (int k = 0; k < n; ++k) {
    const float v = vals[p0 + k];
    const float4* xp = (const float4*)(x + (size_t)cols[p0 + k] * BATCH + lane * 8);
    float4 a = xp[0];
    float4 b = xp[1];
    acc0.x += v * a.x; acc0.y += v * a.y; acc0.z += v * a.z; acc0.w += v * a.w;
    acc1.x += v * b.x; acc1.y += v * b.y; acc1.z += v * b.z; acc1.w += v * b.w;
  }
#endif

  const float bb = bias[row];
  float4* op = (float4*)(out + (size_t)row * BATCH + lane * 8);
  op[0] = make_float4(acc0.x + bb, acc0.y + bb, acc0.z + bb, acc0.w + bb);
  op[1] = make_float4(acc1.x + bb, acc1.y + bb, acc1.z + bb, acc1.w + bb);
}

// ---------------- atomic fallback (if workspace too small) ----------------

__global__ void sk_bias_init(const float* __restrict__ bias, float* __restrict__ out, int nrows) {
  int r = blockIdx.x;
  if (r < nrows) out[(size_t)r * BATCH + threadIdx.x] = bias[r];
}

__global__ void sk_spmm_atomic(const long long* __restrict__ rows,
                               const long long* __restrict__ cols,
                               const float* __restrict__ vals,
                               const float* __restrict__ x,
                               float* __restrict__ out, int nnz) {
  int gid = blockIdx.x * blockDim.x + threadIdx.x;
  int w = gid >> 5, lane = gid & 31;
  if (w >= nnz) return;
  int r = (int)rows[w], c = (int)cols[w];
  float v = vals[w];
  const float4* xp = (const float4*)(x + (size_t)c * BATCH + lane * 8);
  float4 a = xp[0], b = xp[1];
  float* o = out + (size_t)r * BATCH + lane * 8;
  atomicAdd(o + 0, v * a.x); atomicAdd(o + 1, v * a.y);
  atomicAdd(o + 2, v * a.z); atomicAdd(o + 3, v * a.w);
  atomicAdd(o + 4, v * b.x); atomicAdd(o + 5, v * b.y);
  atomicAdd(o + 6, v * b.z); atomicAdd(o + 7, v * b.w);
}

// ---------------- launcher ----------------

extern "C" void kernel_launch(void* const* d_in, const int* in_sizes, int n_in,
                              void* d_out, int out_size, void* d_ws, size_t ws_size,
                              hipStream_t stream) {
  (void)n_in; (void)out_size;
  const float*     x       = (const float*)d_in[0];
  const long long* indices = (const long long*)d_in[1];   // (2, nnz) int64
  const float*     values  = (const float*)d_in[2];
  const float*     bias    = (const float*)d_in[3];
  float*           out     = (float*)d_out;

  const int nnz   = in_sizes[2];
  const int nrows = in_sizes[3];
  const long long* rows  = indices;
  const long long* colsI = indices + nnz;

  // workspace layout (256B aligned chunks)
  char* ws = (char*)d_ws;
  size_t off = 0;
  auto take = [&](size_t bytes) { size_t p = off; off = (off + bytes + 255) & ~(size_t)255; return p; };
  const size_t o_counts = take((size_t)nrows * 4);
  const size_t o_rowptr = take((size_t)(nrows + 1) * 4);
  const size_t o_cursor = take((size_t)nrows * 4);
  const size_t o_cols   = take((size_t)nnz * 4);
  const size_t o_vals   = take((size_t)nnz * 4);

  if (ws_size >= off) {
    int*   counts  = (int*)(ws + o_counts);
    int*   row_ptr = (int*)(ws + o_rowptr);
    int*   cursor  = (int*)(ws + o_cursor);
    int*   colsC   = (int*)(ws + o_cols);
    float* valsC   = (float*)(ws + o_vals);

    sk_zero_i32<<<(nrows + 255) / 256, 256, 0, stream>>>(counts, nrows);
    sk_hist_rows<<<(nnz + 255) / 256, 256, 0, stream>>>(rows, counts, nnz);
    sk_exscan<<<1, 1024, 0, stream>>>(counts, row_ptr, cursor, nrows);
    sk_scatter<<<(nnz + 255) / 256, 256, 0, stream>>>(rows, colsI, values, cursor,
                                                      colsC, valsC, nnz);
    sk_spmm_csr<<<(nrows + WAVES_PER_BLOCK - 1) / WAVES_PER_BLOCK, 256, 0, stream>>>(
        x, row_ptr, colsC, valsC, bias, out, nrows);
  } else {
    sk_bias_init<<<nrows, BATCH, 0, stream>>>(bias, out, nrows);
    int totalThreads32 = nnz * 32;
    sk_spmm_atomic<<<(totalThreads32 + 255) / 256, 256, 0, stream>>>(
        rows, colsI, values, x, out, nnz);
  }
}